// GraphConvLSTMCell_10763188044358
// MI455X (gfx1250) — compile-verified
//
#include <hip/hip_runtime.h>
#include <hip/hip_bf16.h>
#include <math.h>

typedef __attribute__((ext_vector_type(2))) float v2f;
typedef __attribute__((ext_vector_type(8))) float v8f;

#define NNODES 20000
#define HDIM   128
#define IDIM   64
#define BATCH  4

// Fast gate activations: single hardware v_exp_f32, no libm branch diamonds.
// Saturate correctly for |x| large (exp -> 0 or inf).
__device__ __forceinline__ float fast_sigmoid(float x) {
    return 1.0f / (1.0f + __expf(-x));
}
__device__ __forceinline__ float fast_tanh(float x) {
    // tanh(x) = 1 - 2/(1 + e^{2x}); exact limits at +-inf.
    return 1.0f - 2.0f / (1.0f + __expf(2.0f * x));
}

// ---------------------------------------------------------------------------
// 1) zero agg + degree counters (grid-stride over 32-bit words)
// ---------------------------------------------------------------------------
__global__ __launch_bounds__(256) void zero_kernel(unsigned int* ws, long long nwords) {
    long long i = (long long)blockIdx.x * blockDim.x + threadIdx.x;
    long long stride = (long long)gridDim.x * blockDim.x;
    for (; i < nwords; i += stride) ws[i] = 0u;
}

// ---------------------------------------------------------------------------
// 2) degree counts
// ---------------------------------------------------------------------------
__global__ __launch_bounds__(256) void degree_kernel(const int* __restrict__ src,
                                                     const int* __restrict__ dst,
                                                     int* __restrict__ deg_src,
                                                     int* __restrict__ deg_dst, int E) {
    int e = blockIdx.x * blockDim.x + threadIdx.x;
    if (e >= E) return;
    atomicAdd(&deg_src[src[e]], 1);
    atomicAdd(&deg_dst[dst[e]], 1);
}

// ---------------------------------------------------------------------------
// 3) dn = rsqrt(clip(deg, 1))
// ---------------------------------------------------------------------------
__global__ __launch_bounds__(256) void dn_kernel(const int* __restrict__ deg_src,
                                                 const int* __restrict__ deg_dst,
                                                 float* __restrict__ dn_src,
                                                 float* __restrict__ dn_dst) {
    int n = blockIdx.x * blockDim.x + threadIdx.x;
    if (n >= NNODES) return;
    dn_src[n] = rsqrtf(fmaxf((float)deg_src[n], 1.0f));
    dn_dst[n] = rsqrtf(fmaxf((float)deg_dst[n], 1.0f));
}

// ---------------------------------------------------------------------------
// 4) tiny pre-activation GEMMs: xg[gate][b][h] = x[b,:] @ W[:,h] + bias[h]
//    4 gates * 4 batches * 128 = 2048 outputs, one block.
// ---------------------------------------------------------------------------
__global__ __launch_bounds__(256) void xgates_kernel(const float* __restrict__ x,
                                                     const float* __restrict__ Wi, const float* __restrict__ bi,
                                                     const float* __restrict__ Wf, const float* __restrict__ bf,
                                                     const float* __restrict__ Wo, const float* __restrict__ bo,
                                                     const float* __restrict__ Wc, const float* __restrict__ bc,
                                                     float* __restrict__ xg) {
    const float* Ws[4] = {Wi, Wf, Wo, Wc};
    const float* bs[4] = {bi, bf, bo, bc};
    for (int o = threadIdx.x; o < 4 * BATCH * HDIM; o += blockDim.x) {
        int gate = o >> 9;              // /512
        int b    = (o >> 7) & 3;        // /128 % 4
        int h    = o & (HDIM - 1);
        const float* W = Ws[gate];
        float s = bs[gate][h];
        #pragma unroll 8
        for (int i = 0; i < IDIM; ++i) s += x[b * IDIM + i] * W[i * HDIM + h];
        xg[o] = s;
    }
}

// ---------------------------------------------------------------------------
// 5) edge scatter: agg[b, dst[e], h] += h_prev[b, src[e], h] * dn_src[src[e]]
//    tid = (e, h); consecutive lanes share e -> coalesced gather & atomics.
//    ~164M f32 atomics landing in the 41MB L2-resident agg buffer.
// ---------------------------------------------------------------------------
__global__ __launch_bounds__(256) void scatter_kernel(const float* __restrict__ h_prev,
                                                      const int* __restrict__ src,
                                                      const int* __restrict__ dst,
                                                      const float* __restrict__ dn_src,
                                                      float* __restrict__ agg, int E) {
    long long tid = (long long)blockIdx.x * blockDim.x + threadIdx.x;
    long long total = (long long)E * HDIM;
    if (tid >= total) return;
    int e = (int)(tid >> 7);
    int h = (int)(tid & (HDIM - 1));
    int s = src[e], d = dst[e];
    float dn = dn_src[s];
    const long long NH = (long long)NNODES * HDIM;
    long long sOff = (long long)s * HDIM + h;
    long long dOff = (long long)d * HDIM + h;
    #pragma unroll
    for (int b = 0; b < BATCH; ++b) {
        float v = h_prev[b * NH + sOff] * dn;
        atomicAdd(&agg[b * NH + dOff], v);
    }
}

// ---------------------------------------------------------------------------
// 6) fused GEMM (agg * dn_dst) @ Wg + bg -> gh, + LSTM gate epilogue.
//    One wave per 16x16 output tile; 8 waves/block span one 16-row strip.
//    K=128 -> 32x v_wmma_f32_16x16x4_f32 per tile.
//
//    A frag (16x4 f32): lanes 0-15 hold M=lane, K = k+0,k+1 in a.x,a.y;
//                       lanes 16-31 hold M=lane-16, K = k+2,k+3.
//    B frag (4x16 f32): lanes 0-15 hold N=lane, K = k+0 (b.x), k+1 (b.y);
//                       lanes 16-31 hold N=lane-16, K = k+2, k+3.
//    C/D (16x16 f32):   VGPR v, lanes 0-15: M=v, N=lane; lanes 16-31: M=v+8.
// ---------------------------------------------------------------------------
__global__ __launch_bounds__(256) void gemm_lstm_kernel(const float* __restrict__ agg,
                                                        const float* __restrict__ Wg,
                                                        const float* __restrict__ bg,
                                                        const float* __restrict__ dn_dst,
                                                        const float* __restrict__ xg,
                                                        const float* __restrict__ c_prev,
                                                        float* __restrict__ out_h,
                                                        float* __restrict__ out_c) {
    const int lane = threadIdx.x & 31;
    const int wave = threadIdx.x >> 5;          // 0..7 -> column tile
    const int row0 = blockIdx.x * 16;           // 16-row strip; 20000 % 16 == 0 so a
                                                // strip never crosses a batch boundary
    const int colBase = wave * 16;

    const int mLane = lane & 15;                // M (A/C) or N (B) index within tile
    const int kOff  = (lane >> 4) << 1;         // 0 for lanes 0-15, 2 for lanes 16-31

    const int aRow = row0 + mLane;
    const float dn = dn_dst[aRow % NNODES];     // row-constant per lane; fold into A
    const float* Abase = agg + (long long)aRow * HDIM + kOff;
    const int nCol = colBase + mLane;

    // prefetch the next strip's A row (streaming read of agg)
    __builtin_prefetch(Abase + 16 * HDIM, 0, 1);

    v8f acc = {};
    #pragma unroll 4
    for (int k = 0; k < HDIM; k += 4) {
        v2f a = *(const v2f*)(Abase + k);       // A[aRow][k+kOff .. k+kOff+1], 8B aligned
        a.x *= dn; a.y *= dn;
        v2f b;
        b.x = Wg[(k + kOff) * HDIM + nCol];     // B[k+kOff][nCol]
        b.y = Wg[(k + kOff + 1) * HDIM + nCol];
        acc = __builtin_amdgcn_wmma_f32_16x16x4_f32(
            /*neg_a=*/false, a, /*neg_b=*/false, b,
            /*c_mod=*/(short)0, acc, /*reuse_a=*/false, /*reuse_b=*/false);
    }

    // Batch index is constant across the whole 16-row strip.
    const int bIdx  = row0 / NNODES;
    const int nBase = row0 - bIdx * NNODES;
    const int col   = colBase + mLane;

    const float bgc = bg[col];
    const float xi = xg[0 * 512 + bIdx * HDIM + col];
    const float xf = xg[1 * 512 + bIdx * HDIM + col];
    const float xo = xg[2 * 512 + bIdx * HDIM + col];
    const float xc = xg[3 * 512 + bIdx * HDIM + col];

    const long long outBase = ((long long)bIdx * NNODES + nBase) * HDIM + col;

    #pragma unroll
    for (int v = 0; v < 8; ++v) {
        int m = v + ((lane >> 4) << 3);         // D layout: lanes 16-31 carry M = v+8
        float gh = acc[v] + bgc;

        float ig = fast_sigmoid(xi + gh);
        float fg = fast_sigmoid(xf + gh);
        float og = fast_sigmoid(xo + gh);
        float ct = fast_tanh(xc + gh);

        long long idx = outBase + (long long)m * HDIM;
        float cn = fg * c_prev[idx] + ig * ct;
        out_c[idx] = cn;
        out_h[idx] = og * fast_tanh(cn);
    }
}

// ---------------------------------------------------------------------------
// launch
// ---------------------------------------------------------------------------
extern "C" void kernel_launch(void* const* d_in, const int* in_sizes, int n_in,
                              void* d_out, int out_size, void* d_ws, size_t ws_size,
                              hipStream_t stream) {
    const float* x      = (const float*)d_in[0];
    const float* h_prev = (const float*)d_in[1];
    const float* c_prev = (const float*)d_in[2];
    const int*   src    = (const int*)d_in[3];
    const int*   dst    = (const int*)d_in[4];
    const float* Wi = (const float*)d_in[5];  const float* bi = (const float*)d_in[6];
    const float* Wf = (const float*)d_in[7];  const float* bf = (const float*)d_in[8];
    const float* Wo = (const float*)d_in[9];  const float* bo = (const float*)d_in[10];
    const float* Wc = (const float*)d_in[11]; const float* bc = (const float*)d_in[12];
    const float* Wg = (const float*)d_in[13]; const float* bg = (const float*)d_in[14];
    const int E = in_sizes[3];

    // workspace layout (floats):
    //   agg     : BATCH*NNODES*HDIM = 10,240,000
    //   deg_src : 20000 (int)
    //   deg_dst : 20000 (int)
    //   dn_src  : 20000
    //   dn_dst  : 20000
    //   xg      : 4*BATCH*HDIM = 2048        (total ~41.3 MB)
    float* ws      = (float*)d_ws;
    float* agg     = ws;
    const long long AGG_W = (long long)BATCH * NNODES * HDIM;
    int*   deg_src = (int*)(ws + AGG_W);
    int*   deg_dst = deg_src + NNODES;
    float* dn_src  = (float*)(deg_dst + NNODES);
    float* dn_dst  = dn_src + NNODES;
    float* xg      = dn_dst + NNODES;

    float* out_h = (float*)d_out;
    float* out_c = out_h + AGG_W;

    // 1) zero agg + degree counters
    long long zwords = AGG_W + 2 * NNODES;
    zero_kernel<<<2048, 256, 0, stream>>>((unsigned int*)d_ws, zwords);

    // 2) degrees
    degree_kernel<<<(E + 255) / 256, 256, 0, stream>>>(src, dst, deg_src, deg_dst, E);

    // 3) normalizers
    dn_kernel<<<(NNODES + 255) / 256, 256, 0, stream>>>(deg_src, deg_dst, dn_src, dn_dst);

    // 4) input-gate pre-activations
    xgates_kernel<<<1, 256, 0, stream>>>(x, Wi, bi, Wf, bf, Wo, bo, Wc, bc, xg);

    // 5) edge scatter
    long long swork = (long long)E * HDIM;
    int sblocks = (int)((swork + 255) / 256);
    scatter_kernel<<<sblocks, 256, 0, stream>>>(h_prev, src, dst, dn_src, agg, E);

    // 6) fused WMMA GEMM + LSTM epilogue: 5000 strips of 16 rows, 8 waves/block
    gemm_lstm_kernel<<<(BATCH * NNODES) / 16, 256, 0, stream>>>(
        agg, Wg, bg, dn_dst, xg, c_prev, out_h, out_c);
}